// ToRGB_wavelet_1563368096150
// MI455X (gfx1250) — compile-verified
//
#include <hip/hip_runtime.h>

typedef float v2f __attribute__((ext_vector_type(2)));
typedef float v8f __attribute__((ext_vector_type(8)));

#define C_IN  256
#define C_OUT 12
#define HW    16384   // 128*128

// ---------------------------------------------------------------------------
// Kernel 1: out[b,m,p] = sum_k w[m,k] * in[b,k,p] + bias[m]
// Per-wave: one 16(M) x 64(pixels) tile as 4 accumulators, K chained through
// 64 x V_WMMA_F32_16X16X4_F32 per accumulator group.
// A-frag layout (16x4 f32): lanes 0-15 -> K={0,1}, lanes 16-31 -> K={2,3} (khalf=0/2).
// B-frag layout (4x16 f32): mirrored (rows K across VGPR/lane-half, N across lanes).
// ---------------------------------------------------------------------------
__global__ __launch_bounds__(256) void torgb_wmma_kernel(
    const float* __restrict__ in, const float* __restrict__ w,
    const float* __restrict__ bias, float* __restrict__ out)
{
    __shared__ float wlds[16 * C_IN];   // weight padded to M=16, zero rows 12..15
    for (int i = threadIdx.x; i < 16 * C_IN; i += 256) {
        int m = i >> 8, k = i & 255;
        wlds[i] = (m < C_OUT) ? w[m * C_IN + k] : 0.0f;
    }
    __syncthreads();

    const int lane  = threadIdx.x & 31;
    const int wv    = threadIdx.x >> 5;
    const int gw    = blockIdx.x * 8 + wv;   // global wave id; 4096 total
    const int b     = gw >> 8;               // 256 wave-tiles per batch
    const int p0    = (gw & 255) << 6;       // 64 pixels per wave

    const int mrow  = lane & 15;
    const int khalf = (lane >> 4) << 1;      // 0 or 2
    const int n     = lane & 15;

    const float* pB = in + ((size_t)b * C_IN + khalf) * HW + p0 + n;
    const float* pA = wlds + mrow * C_IN + khalf;

    v8f acc0 = {}, acc1 = {}, acc2 = {}, acc3 = {};
    for (int k0 = 0; k0 < C_IN; k0 += 4) {
        v2f a;
        a.x = pA[k0];
        a.y = pA[k0 + 1];
        const float* pk = pB + (size_t)k0 * HW;
        v2f b0, b1, b2, b3;
        b0.x = pk[0];  b0.y = pk[HW + 0];
        b1.x = pk[16]; b1.y = pk[HW + 16];
        b2.x = pk[32]; b2.y = pk[HW + 32];
        b3.x = pk[48]; b3.y = pk[HW + 48];
        acc0 = __builtin_amdgcn_wmma_f32_16x16x4_f32(false, a, false, b0, (short)0, acc0, false, false);
        acc1 = __builtin_amdgcn_wmma_f32_16x16x4_f32(false, a, false, b1, (short)0, acc1, false, false);
        acc2 = __builtin_amdgcn_wmma_f32_16x16x4_f32(false, a, false, b2, (short)0, acc2, false, false);
        acc3 = __builtin_amdgcn_wmma_f32_16x16x4_f32(false, a, false, b3, (short)0, acc3, false, false);
    }

    // C/D layout: lane L, VGPR v -> M = v + 8*(L>>4), N = L&15
    const int mbase = (lane >> 4) << 3;
    #pragma unroll
    for (int v = 0; v < 8; ++v) {
        int m = mbase + v;
        if (m < C_OUT) {
            float bv = bias[m];
            float* po = out + ((size_t)b * C_OUT + m) * HW + p0 + n;
            po[0]  = acc0[v] + bv;
            po[16] = acc1[v] + bv;
            po[32] = acc2[v] + bv;
            po[48] = acc3[v] + bv;
        }
    }
}

// ---------------------------------------------------------------------------
// Kernel 2: fused dwt(upsample(iwt(skip))) added into out.
//   iwt collapses to a per-pixel Haar mix of skip at [i>>1][j>>1];
//   upsample+dwt collapse to a separable 3x3 conv over t1 with taps
//   f_l = s*[0.25, 1.5, 0.25], f_h = s*[0.25, 0, -0.25], s = 1/sqrt(2).
// ---------------------------------------------------------------------------
__device__ __forceinline__ float t1val(const float* __restrict__ skip,
                                       int b, int c, int i, int j)
{
    if ((unsigned)i >= 128u || (unsigned)j >= 128u) return 0.0f;
    int r = i >> 1, s = j >> 1;
    const float* p = skip + ((size_t)b * 12 + c) * 4096 + r * 64 + s;
    float LL = p[0 * 3 * 4096];
    float LH = p[1 * 3 * 4096];
    float HL = p[2 * 3 * 4096];
    float HH = p[3 * 3 * 4096];
    float sp = (i & 1) ? 1.0f : -1.0f;
    float sq = (j & 1) ? 1.0f : -1.0f;
    return 0.5f * (LL - sp * LH - sq * HL + sp * sq * HH);
}

__global__ __launch_bounds__(256) void wavelet_skip_add_kernel(
    const float* __restrict__ skip, float* __restrict__ out)
{
    const int total = 16 * C_OUT * HW;
    int idx = blockIdx.x * 256 + threadIdx.x;
    if (idx >= total) return;

    int x = idx & 127;
    int y = (idx >> 7) & 127;
    int m = (idx >> 14) % C_OUT;
    int b = idx / (C_OUT * HW);
    int c = m % 3;
    int g = m / 3;          // 0=ll 1=lh 2=hl 3=hh

    const float S = 0.70710678118654752f;
    const float FL[3] = {0.25f * S, 1.50f * S,  0.25f * S};
    const float FH[3] = {0.25f * S, 0.00f,     -0.25f * S};
    const float* fy = (g & 1) ? FH : FL;   // lh/hh use h along rows
    const float* fx = (g & 2) ? FH : FL;   // hl/hh use h along cols

    float acc = 0.0f;
    #pragma unroll
    for (int dr = -1; dr <= 1; ++dr) {
        float cy = fy[dr + 1];
        #pragma unroll
        for (int dc = -1; dc <= 1; ++dc) {
            float cf = cy * fx[dc + 1];
            if (cf != 0.0f)
                acc += cf * t1val(skip, b, c, y + dr, x + dc);
        }
    }
    out[idx] += acc;
}

// ---------------------------------------------------------------------------
extern "C" void kernel_launch(void* const* d_in, const int* in_sizes, int n_in,
                              void* d_out, int out_size, void* d_ws, size_t ws_size,
                              hipStream_t stream)
{
    const float* input  = (const float*)d_in[0];  // [16,256,128,128]
    const float* skip   = (const float*)d_in[1];  // [16,12,64,64]
    const float* weight = (const float*)d_in[2];  // [12,256]
    const float* bias   = (const float*)d_in[3];  // [12]
    float* out = (float*)d_out;                   // [16,12,128,128]

    // 16 batches * 256 wave-tiles / 8 waves-per-block = 512 blocks
    torgb_wmma_kernel<<<512, 256, 0, stream>>>(input, weight, bias, out);

    const int total = 16 * C_OUT * HW;            // 3,145,728
    wavelet_skip_add_kernel<<<(total + 255) / 256, 256, 0, stream>>>(skip, out);
}